// Reservoir2_80075370266952
// MI455X (gfx1250) — compile-verified
//
#include <hip/hip_runtime.h>
#include <math.h>
#include <stdint.h>

#define N_V   256
#define DRR   512
#define LT    2048
#define EMB   100
#define RHO   0.9f
#define DELTA 0.5f
#define BCONST 0.1f
#define BN_EPS 1e-5f

typedef float v2f __attribute__((ext_vector_type(2)));
typedef float v8f __attribute__((ext_vector_type(8)));

__device__ __forceinline__ float warp_allreduce(float v) {
  v += __shfl_xor(v, 16); v += __shfl_xor(v, 8); v += __shfl_xor(v, 4);
  v += __shfl_xor(v, 2);  v += __shfl_xor(v, 1);
  return v;
}

__device__ __forceinline__ float urand(unsigned s) {
  s = s * 747796405u + 2891336453u;
  unsigned w = ((s >> ((s >> 28) + 4u)) ^ s) * 277803737u;
  w = (w >> 22) ^ w;
  return (float)(w >> 8) * (1.0f / 16777216.0f);
}

// ---------------- Phase 1: output1[n][t] = R_state1[t,n,:] . W_out1[n,:]  (HBM-bound, 1.07 GB stream)
__global__ void k_readout1(const float* __restrict__ R1,
                           const float* __restrict__ Wout1,
                           float* __restrict__ out1) {
  int wid  = blockIdx.x * 8 + (threadIdx.x >> 5);
  int lane = threadIdx.x & 31;
  int t = wid >> 8;
  int n = wid & 255;
  const float4* rp = (const float4*)(R1 + ((long)t * N_V + n) * DRR);
  const float4* wp = (const float4*)(Wout1 + (long)n * DRR);
  float acc = 0.f;
#pragma unroll
  for (int j = 0; j < 4; ++j) {
    float4 a = rp[lane + j * 32];
    float4 b = wp[lane + j * 32];
    acc += a.x * b.x + a.y * b.y + a.z * b.z + a.w * b.w;
  }
  acc = warp_allreduce(acc);
  if (lane == 0) out1[(long)n * LT + t] = acc;
}

// ---------------- Phase 2: x = relu(out1 @ fc_W^T + fc_b)   (256x100, K=2048)
__global__ void k_fc1(const float* __restrict__ out1, const float* __restrict__ fcW,
                      const float* __restrict__ fcb, float* __restrict__ x) {
  int wid  = blockIdx.x * 8 + (threadIdx.x >> 5);
  int lane = threadIdx.x & 31;
  int n = wid / EMB, e = wid - n * EMB;
  if (n >= N_V) return;
  const float4* ap = (const float4*)(out1 + (long)n * LT);
  const float4* bp = (const float4*)(fcW + (long)e * LT);
  float acc = 0.f;
#pragma unroll
  for (int j = 0; j < 16; ++j) {
    float4 a = ap[lane + j * 32];
    float4 b = bp[lane + j * 32];
    acc += a.x * b.x + a.y * b.y + a.z * b.z + a.w * b.w;
  }
  acc = warp_allreduce(acc);
  if (lane == 0) {
    float v = acc + fcb[e];
    x[n * EMB + e] = v > 0.f ? v : 0.f;
  }
}

// ---------------- Phase 3: BatchNorm (training stats over 256 rows, biased var)
__global__ void k_bn(const float* __restrict__ x, const float* __restrict__ gamma,
                     const float* __restrict__ beta, float* __restrict__ xn) {
  int e = blockIdx.x;
  int n = threadIdx.x;
  float v = x[n * EMB + e];
  __shared__ float s1[256], s2[256];
  s1[n] = v; s2[n] = v * v;
  __syncthreads();
  for (int off = 128; off > 0; off >>= 1) {
    if (n < off) { s1[n] += s1[n + off]; s2[n] += s2[n + off]; }
    __syncthreads();
  }
  __shared__ float sc, sh;
  if (n == 0) {
    float mean = s1[0] * (1.f / 256.f);
    float var  = s2[0] * (1.f / 256.f) - mean * mean;
    float s    = gamma[e] * rsqrtf(var + BN_EPS);
    sc = s; sh = beta[e] - mean * s;
  }
  __syncthreads();
  xn[n * EMB + e] = v * sc + sh;
}

// ---------------- Phase 4: edge MLP + gumbel argmax -> A^T (zero diag), one wave per edge
__global__ void k_edges(const float* __restrict__ xn, const float* __restrict__ Wo,
                        const float* __restrict__ bo, const float* __restrict__ catW,
                        const float* __restrict__ catb, float* __restrict__ ATws) {
  int e    = blockIdx.x * 8 + (threadIdx.x >> 5);
  int lane = threadIdx.x & 31;
  int i = e >> 8;   // receiver
  int j = e & 255;  // sender
  const float* xs = xn + j * EMB;
  const float* xr = xn + i * EMB;
  int h0 = lane, h1 = lane + 32, h2 = lane + 64, h3 = lane + 96;
  float a0 = 0.f, a1 = 0.f, a2 = 0.f, a3 = 0.f;
  for (int k = 0; k < EMB; ++k) {
    float s = xs[k], t = xr[k];
    a0 += Wo[h0 * 200 + k] * s + Wo[h0 * 200 + 100 + k] * t;
    a1 += Wo[h1 * 200 + k] * s + Wo[h1 * 200 + 100 + k] * t;
    a2 += Wo[h2 * 200 + k] * s + Wo[h2 * 200 + 100 + k] * t;
    if (h3 < EMB) a3 += Wo[h3 * 200 + k] * s + Wo[h3 * 200 + 100 + k] * t;
  }
  float e0 = fmaxf(a0 + bo[h0], 0.f);
  float e1 = fmaxf(a1 + bo[h1], 0.f);
  float e2 = fmaxf(a2 + bo[h2], 0.f);
  float e3 = (h3 < EMB) ? fmaxf(a3 + bo[h3], 0.f) : 0.f;
  float l0 = catW[h0] * e0 + catW[h1] * e1 + catW[h2] * e2 + ((h3 < EMB) ? catW[h3] * e3 : 0.f);
  float l1 = catW[100 + h0] * e0 + catW[100 + h1] * e1 + catW[100 + h2] * e2 +
             ((h3 < EMB) ? catW[100 + h3] * e3 : 0.f);
  l0 = warp_allreduce(l0) + catb[0];
  l1 = warp_allreduce(l1) + catb[1];
  float u0 = urand(2u * (unsigned)e);
  float u1 = urand(2u * (unsigned)e + 1u);
  float g0 = -logf(-logf(u0 + 1e-20f) + 1e-20f);
  float g1 = -logf(-logf(u1 + 1e-20f) + 1e-20f);
  if (lane == 0) {
    float y0 = ((l0 + g0) >= (l1 + g1)) ? 1.f : 0.f;
    ATws[j * 256 + i] = (i == j) ? 0.f : y0;   // store A transposed directly
  }
}

// ---------------- init: R0 = 0, output row 0 = 0
__global__ void k_init(float* __restrict__ R0, float* __restrict__ out0) {
  int idx = blockIdx.x * 256 + threadIdx.x;
  if (idx < N_V * DRR) R0[idx] = 0.f;
  if (idx < N_V) out0[idx] = 0.f;
}

// ---------------- Phase 5: fused scan step, one launch per step.
// R_new = tanh(RHO*((AT@R_prev)@Rnet) + DELTA*u⊗win + B) via associativity.
// 16 WGs x 512 threads; WG `panel` owns 16 output rows end-to-end:
//   Phase A: P = AT_panel @ R_prev -> LDS   (fp32 WMMA 16x16x4)
//   Phase B: P @ Rnet -> tanh -> store + fused W_out2 readout (shfl+LDS reduce).
// AT panel is staged into LDS with GLOBAL_LOAD_ASYNC_TO_LDS_B128 (ASYNCcnt-tracked
// DMA, no VGPR round-trip), completed with s_wait_asynccnt before the barrier.
__global__ __launch_bounds__(512) void k_scan(
    const float* __restrict__ Rprev, float* __restrict__ Rnew,
    const float* __restrict__ AT, const float* __restrict__ Rnet,
    const float* __restrict__ u, const float* __restrict__ win,
    const float* __restrict__ Wout2, float* __restrict__ outrow) {
  constexpr int LDA = 260;   // 256 + 4 pad (keeps rows 16B-aligned for b128 async stores)
  constexpr int LDP = 514;   // 512 + 2 pad (even -> b64-aligned v2f loads)
  __shared__ float Ald[16 * LDA];
  __shared__ float Pld[16 * LDP];
  __shared__ float red[16 * 17];
  const int tid   = threadIdx.x;
  const int w     = tid >> 5;
  const int lane  = tid & 31;
  const int r     = lane & 15;
  const int hi    = lane >> 4;
  const int panel = blockIdx.x;

  // async-stage AT panel (16 x 256 floats = 1024 x 16B) straight into LDS
#pragma unroll
  for (int it = 0; it < 2; ++it) {
    int idx = tid + it * 512;              // float4 index
    int rr  = idx >> 6;                    // row (64 float4 per row)
    int c4  = idx & 63;
    unsigned lds = (unsigned)(uintptr_t)(void*)&Ald[rr * LDA + c4 * 4];
    const float* g = AT + (panel * 16 + rr) * 256 + c4 * 4;
    asm volatile("global_load_async_to_lds_b128 %0, %1, off"
                 :: "v"(lds), "v"(g) : "memory");
  }
  asm volatile("s_wait_asynccnt 0x0" ::: "memory");
  __syncthreads();

  const int n0 = w * 32;  // this wave owns output cols [n0,n0+16) and [n0+16,n0+32)

  // ---- Phase A: P(16x512) = AT_panel(16x256) @ Rprev(256x512), K=256
  {
    v8f c0 = {}; v8f c1 = {};
    const float* ap = Ald + r * LDA + 2 * hi;
    for (int kk = 0; kk < 256; kk += 4) {
      v2f a = *(const v2f*)(ap + kk);                 // A[r][kk+2hi .. +1]
      const float* bp = Rprev + (kk + 2 * hi) * DRR + n0 + r;
      v2f b0; b0.x = bp[0];  b0.y = bp[DRR];
      v2f b1; b1.x = bp[16]; b1.y = bp[DRR + 16];
      c0 = __builtin_amdgcn_wmma_f32_16x16x4_f32(false, a, false, b0, (short)0, c0, false, false);
      c1 = __builtin_amdgcn_wmma_f32_16x16x4_f32(false, a, false, b1, (short)0, c1, false, false);
    }
    const int m = 8 * hi;
#pragma unroll
    for (int v = 0; v < 8; ++v) {
      Pld[(m + v) * LDP + n0 + r]      = c0[v];
      Pld[(m + v) * LDP + n0 + 16 + r] = c1[v];
    }
  }
  __syncthreads();

  // ---- Phase B: C(16x512) = P @ Rnet, K=512, then tanh epilogue + fused readout
  {
    v8f c0 = {}; v8f c1 = {};
    const float* ap = Pld + r * LDP + 2 * hi;
    for (int kk = 0; kk < 512; kk += 4) {
      v2f a = *(const v2f*)(ap + kk);
      const float* bp = Rnet + (kk + 2 * hi) * DRR + n0 + r;
      v2f b0; b0.x = bp[0];  b0.y = bp[DRR];
      v2f b1; b1.x = bp[16]; b1.y = bp[DRR + 16];
      c0 = __builtin_amdgcn_wmma_f32_16x16x4_f32(false, a, false, b0, (short)0, c0, false, false);
      c1 = __builtin_amdgcn_wmma_f32_16x16x4_f32(false, a, false, b1, (short)0, c1, false, false);
    }
    float pr[8];
    const int col0 = n0 + r, col1 = n0 + 16 + r;
    const float win0 = win[col0], win1 = win[col1];
#pragma unroll
    for (int v = 0; v < 8; ++v) {
      int node = panel * 16 + 8 * hi + v;
      float uu = DELTA * u[node];
      float v0 = tanhf(RHO * c0[v] + uu * win0 + BCONST);
      float v1 = tanhf(RHO * c1[v] + uu * win1 + BCONST);
      Rnew[node * DRR + col0] = v0;
      Rnew[node * DRR + col1] = v1;
      pr[v] = v0 * Wout2[node * DRR + col0] + v1 * Wout2[node * DRR + col1];
    }
#pragma unroll
    for (int v = 0; v < 8; ++v) {          // reduce over the 16 lanes of each half
      float s = pr[v];
      s += __shfl_xor(s, 1); s += __shfl_xor(s, 2);
      s += __shfl_xor(s, 4); s += __shfl_xor(s, 8);
      pr[v] = s;
    }
#pragma unroll
    for (int v = 0; v < 8; ++v)
      if (r == v) red[(8 * hi + v) * 17 + w] = pr[v];
  }
  __syncthreads();
  if (tid < 16) {
    float s = 0.f;
#pragma unroll
    for (int ww = 0; ww < 16; ++ww) s += red[tid * 17 + ww];
    outrow[panel * 16 + tid] = s;           // fused W_out2 readout, no atomics
  }
}

extern "C" void kernel_launch(void* const* d_in, const int* in_sizes, int n_in,
                              void* d_out, int out_size, void* d_ws, size_t ws_size,
                              hipStream_t stream) {
  const float* R_state1 = (const float*)d_in[0];
  const float* train    = (const float*)d_in[1];
  const float* Wout1    = (const float*)d_in[2];
  const float* Wout2    = (const float*)d_in[3];
  const float* Win      = (const float*)d_in[4];
  const float* Rnet     = (const float*)d_in[5];
  const float* fcW      = (const float*)d_in[6];
  const float* fcb      = (const float*)d_in[7];
  const float* gamma    = (const float*)d_in[8];
  const float* beta     = (const float*)d_in[9];
  const float* foW      = (const float*)d_in[10];
  const float* fob      = (const float*)d_in[11];
  const float* fcatW    = (const float*)d_in[12];
  const float* fcatb    = (const float*)d_in[13];
  float* out = (float*)d_out;

  float* ws   = (float*)d_ws;
  float* out1 = ws;                                  // 256*2048
  float* xbuf = out1 + (size_t)N_V * LT;             // 256*100
  float* xn   = xbuf + N_V * EMB;                    // 256*100
  float* ATws = xn + N_V * EMB;                      // 256*256
  float* Ra   = ATws + N_V * N_V;                    // 256*512
  float* Rb   = Ra + N_V * DRR;                      // 256*512

  k_readout1<<<65536, 256, 0, stream>>>(R_state1, Wout1, out1);
  k_fc1<<<3200, 256, 0, stream>>>(out1, fcW, fcb, xbuf);
  k_bn<<<EMB, 256, 0, stream>>>(xbuf, gamma, beta, xn);
  k_edges<<<8192, 256, 0, stream>>>(xn, foW, fob, fcatW, fcatb, ATws);
  k_init<<<512, 256, 0, stream>>>(Ra, out);

  const float* Rp = Ra;
  float* Rn = Rb;
  for (int s = 1; s < LT; ++s) {
    k_scan<<<16, 512, 0, stream>>>(Rp, Rn, ATws, Rnet,
                                   train + (size_t)s * N_V, Win, Wout2,
                                   out + (size_t)s * N_V);
    const float* t = Rn; Rn = (float*)Rp; Rp = t;
  }
}